// QueryAndGroup_75093208203665
// MI455X (gfx1250) — compile-verified
//
#include <hip/hip_runtime.h>

typedef float v2f __attribute__((ext_vector_type(2)));
typedef float v8f __attribute__((ext_vector_type(8)));
typedef int   v4i_b __attribute__((vector_size(16)));   // matches builtin param type

#define B_    8
#define N_    8192
#define P_    2048
#define C_    64
#define S_    32
#define R2_   0.01f
#define WAVES 4
#define CHUNK 512                  // points staged per block per step (6 KB)
#define NCHUNK (N_ / CHUNK)        // 16

#if __has_builtin(__builtin_amdgcn_global_load_async_to_lds_b128)
#define USE_ASYNC_LDS 1
#else
#define USE_ASYNC_LDS 0
#endif

static __device__ __forceinline__ void wait_async0() {
#if USE_ASYNC_LDS
#if __has_builtin(__builtin_amdgcn_s_wait_asynccnt)
    __builtin_amdgcn_s_wait_asynccnt(0);
#else
    asm volatile("s_wait_asynccnt 0x0" ::: "memory");
#endif
#endif
}

__global__ __launch_bounds__(WAVES * 32) void ballquery_group_wmma(
    const float* __restrict__ xyz,      // (B, N, 3)
    const float* __restrict__ new_xyz,  // (B, P, 3)
    const float* __restrict__ feat,     // (B, C, N)
    float* __restrict__ out)            // (B, 3 + C, P, S)
{
    // per-lane top-32 candidate lists (stride 33 -> conflict-free banks)
    __shared__ float s_d2 [WAVES * 32 * 33];
    __shared__ int   s_idx[WAVES * 32 * 33];
    __shared__ int   s_cnt[WAVES * 32];
    __shared__ int   s_fin[WAVES * 16 * 32];
    // double-buffered point staging: 2 x 512 points x 3 floats = 12 KB
    __shared__ __align__(16) float s_pts[2][CHUNK * 3];

    const int tid   = threadIdx.x;
    const int lane  = tid & 31;
    const int w     = tid >> 5;
    const int b     = blockIdx.y;
    const int pbase = (blockIdx.x * WAVES + w) * 16;
    const int col   = lane & 15;          // center column owned by this lane pair
    const bool hi   = lane >= 16;
    const int p     = pbase + col;

    const float* xb = xyz + (size_t)b * N_ * 3;

    // ---- B operand: center column [-2cx, -2cy, -2cz, 1]
    //      (lanes<16 hold rows K=0,1; lanes>=16 hold rows K=2,3)
    const float cx = new_xyz[((size_t)b * P_ + p) * 3 + 0];
    const float cy = new_xyz[((size_t)b * P_ + p) * 3 + 1];
    const float cz = new_xyz[((size_t)b * P_ + p) * 3 + 2];
    const float cnorm = cx * cx + cy * cy + cz * cz;
    v2f bvec;
    bvec.x = hi ? (-2.0f * cz) : (-2.0f * cx);
    bvec.y = hi ? 1.0f         : (-2.0f * cy);

    const int slotBase = (w * 32 + lane) * 33;
    int   cnt = 0, nsel = 0, maxslot = 0;
    float maxd = -1.0f;

    // ---- stage chunk 0
    {
#if USE_ASYNC_LDS
        v4i_b* gsrc = (v4i_b*)(xb);
        v4i_b* ldst = (v4i_b*)&s_pts[0][0];
        #pragma unroll
        for (int i = 0; i < 3; ++i) {
            const int e = i * 128 + tid;
            __builtin_amdgcn_global_load_async_to_lds_b128(gsrc + e, ldst + e, 0, 0);
        }
#else
        const float4* gsrc = (const float4*)(xb);
        #pragma unroll
        for (int i = 0; i < 3; ++i) {
            const int e = i * 128 + tid;
            *(float4*)&s_pts[0][e * 4] = gsrc[e];
        }
#endif
    }

    // ---- distance phase: async double-buffered chunks, WMMA d2 tiles
    for (int ch = 0; ch < NCHUNK; ++ch) {
        wait_async0();          // my chunk-ch async copies done
        __syncthreads();        // everyone's copies visible; prev readers drained

        if (ch + 1 < NCHUNK) {  // stream next chunk while computing this one
#if USE_ASYNC_LDS
            v4i_b* gsrc = (v4i_b*)(xb + (size_t)(ch + 1) * CHUNK * 3);
            v4i_b* ldst = (v4i_b*)&s_pts[(ch + 1) & 1][0];
            #pragma unroll
            for (int i = 0; i < 3; ++i) {
                const int e = i * 128 + tid;
                __builtin_amdgcn_global_load_async_to_lds_b128(gsrc + e, ldst + e, 0, 0);
            }
#else
            const float4* gsrc = (const float4*)(xb + (size_t)(ch + 1) * CHUNK * 3);
            #pragma unroll
            for (int i = 0; i < 3; ++i) {
                const int e = i * 128 + tid;
                *(float4*)&s_pts[(ch + 1) & 1][e * 4] = gsrc[e];
            }
#endif
        }
        if (ch + 2 < NCHUNK)    // keep chunk+2 warming in L2
            __builtin_prefetch(xb + (size_t)(ch + 2) * CHUNK * 3, 0, 1);

        const float* pts = &s_pts[ch & 1][0];
        for (int t = 0; t < CHUNK / 16; ++t) {
            const int j  = t * 16 + col;
            const float px = pts[j * 3 + 0];
            const float py = pts[j * 3 + 1];
            const float pz = pts[j * 3 + 2];
            const float pn = px * px + py * py + pz * pz;
            // A operand: point row [px, py, pz, ||p||^2] (lo: K=0,1; hi: K=2,3)
            v2f avec;
            avec.x = hi ? pz : px;
            avec.y = hi ? pn : py;

            v8f c = {};
            c = __builtin_amdgcn_wmma_f32_16x16x4_f32(
                    false, avec, false, bvec, (short)0, c, false, false);

            const int nbase = ch * CHUNK + t * 16;
            // C[k] = ||p||^2 - 2 p.c for point row m = k + (hi?8:0), center = col
            #pragma unroll
            for (int k = 0; k < 8; ++k) {
                const float d2 = c[k] + cnorm;
                if (d2 <= R2_) {
                    const int id = nbase + k + (hi ? 8 : 0);
                    ++cnt;
                    if (nsel < S_) {
                        s_d2 [slotBase + nsel] = d2;
                        s_idx[slotBase + nsel] = id;
                        if (d2 > maxd) { maxd = d2; maxslot = nsel; }
                        ++nsel;
                    } else if (d2 < maxd) {   // strict: ties keep lower index
                        s_d2 [slotBase + maxslot] = d2;
                        s_idx[slotBase + maxslot] = id;
                        maxd = -1.0f;
                        for (int q = 0; q < S_; ++q) {
                            const float dt = s_d2[slotBase + q];
                            if (dt > maxd) { maxd = dt; maxslot = q; }
                        }
                    }
                }
            }
        }
    }
    s_cnt[w * 32 + lane] = cnt;
    __syncthreads();

    // ---- finalize: lane pair (L, L+16) merged by lane L
    if (lane < 16) {
        const int loBase = (w * 32 + lane) * 33;
        const int hiBase = (w * 32 + lane + 16) * 33;
        const int cLo = s_cnt[w * 32 + lane];
        const int cHi = s_cnt[w * 32 + lane + 16];
        const int total = cLo + cHi;
        const int fb = (w * 16 + lane) * 32;

        if (total >= S_) {
            // 32 nearest by (d2, idx) ascending — matches top_k ordering
            const int nLo = cLo < S_ ? cLo : S_;
            const int nHi = cHi < S_ ? cHi : S_;
            for (int s = 0; s < S_; ++s) {
                float best = 3.0e38f; int bestId = 0x7fffffff; int bestSlot = loBase;
                for (int t = 0; t < S_; ++t) {
                    if (t < nLo) {
                        const float d = s_d2[loBase + t];
                        const int  id = s_idx[loBase + t];
                        if (d < best || (d == best && id < bestId)) { best = d; bestId = id; bestSlot = loBase + t; }
                    }
                    if (t < nHi) {
                        const float d = s_d2[hiBase + t];
                        const int  id = s_idx[hiBase + t];
                        if (d < best || (d == best && id < bestId)) { best = d; bestId = id; bestSlot = hiBase + t; }
                    }
                }
                s_fin[fb + s] = bestId;
                s_d2[bestSlot] = 3.0e38f;   // evict
            }
        } else {
            // fewer than 32 valid: both lists are idx-ascending (append-only);
            // 2-way merge, pad with last valid (0 if none)
            int i = 0, j = 0, last = 0;
            for (int s = 0; s < S_; ++s) {
                int v;
                if (i < cLo && (j >= cHi || s_idx[loBase + i] < s_idx[hiBase + j])) v = s_idx[loBase + i++];
                else if (j < cHi)                                                   v = s_idx[hiBase + j++];
                else                                                                v = last;
                last = v;
                s_fin[fb + s] = v;
            }
        }
    }
    __syncthreads();

    // ---- gather phase: lane = sample index -> coalesced 128B stores
    const float* fbb = feat + (size_t)b * C_ * N_;
    float*       ob  = out  + (size_t)b * (3 + C_) * P_ * S_;
    for (int t = 0; t < 16; ++t) {
        const int pp = pbase + t;
        const int id = s_fin[(w * 16 + t) * 32 + lane];
        const float gx = xb[(size_t)id * 3 + 0] - new_xyz[((size_t)b * P_ + pp) * 3 + 0];
        const float gy = xb[(size_t)id * 3 + 1] - new_xyz[((size_t)b * P_ + pp) * 3 + 1];
        const float gz = xb[(size_t)id * 3 + 2] - new_xyz[((size_t)b * P_ + pp) * 3 + 2];
        ob[((size_t)0 * P_ + pp) * S_ + lane] = gx;
        ob[((size_t)1 * P_ + pp) * S_ + lane] = gy;
        ob[((size_t)2 * P_ + pp) * S_ + lane] = gz;
        #pragma unroll 4
        for (int c = 0; c < C_; ++c) {
            ob[((size_t)(3 + c) * P_ + pp) * S_ + lane] = fbb[(size_t)c * N_ + id];
        }
    }
}

extern "C" void kernel_launch(void* const* d_in, const int* in_sizes, int n_in,
                              void* d_out, int out_size, void* d_ws, size_t ws_size,
                              hipStream_t stream) {
    const float* xyz     = (const float*)d_in[0];
    const float* new_xyz = (const float*)d_in[1];
    const float* feat    = (const float*)d_in[2];
    float*       out     = (float*)d_out;
    (void)in_sizes; (void)n_in; (void)out_size; (void)d_ws; (void)ws_size;

    dim3 grid(P_ / (WAVES * 16), B_);   // (32, 8)
    dim3 block(WAVES * 32);             // 128 threads = 4 waves, 16 centers each
    hipLaunchKernelGGL(ballquery_group_wmma, grid, block, 0, stream,
                       xyz, new_xyz, feat, out);
}